// MTDecoder_71914932404310
// MI455X (gfx1250) — compile-verified
//
#include <hip/hip_runtime.h>
#include <math.h>

typedef __attribute__((ext_vector_type(2))) float v2f;
typedef __attribute__((ext_vector_type(8))) float v8f;

#define B_   32
#define H_   512
#define V_   32000
#define G_   2048      // 4*H
#define TMAX 64

__device__ __forceinline__ float sigmoidf_(float x) {
    return 1.0f / (1.0f + expf(-x));
}

// ---------------- init: h0/c0/h1/c1 = 0, tok = START_INDEX ----------------
__global__ void init_state(float* h0, float* c0, float* h1, float* c1, int* tok) {
    int i = blockIdx.x * blockDim.x + threadIdx.x;
    if (i < B_ * H_) { h0[i] = 0.f; c0[i] = 0.f; h1[i] = 0.f; c1[i] = 0.f; }
    if (i < B_) tok[i] = 1;
}

// ---------------- x0 = [emb[tok] | context]  (32 x 1024) ----------------
__global__ void embed_concat(const float* __restrict__ emb, const float* __restrict__ ctx,
                             const int* __restrict__ tok, float* __restrict__ x0) {
    int i = blockIdx.x * blockDim.x + threadIdx.x;   // 0 .. 32*1024-1
    int m = i >> 10, j = i & 1023;
    x0[i] = (j < 512) ? emb[(long)tok[m] * 512 + j] : ctx[m * 512 + (j - 512)];
}

// ---------------- out[32 x N] = A1*W1^T + A2*W2^T + b1 (+b2) ----------------
// A row-major [32 x K], W row-major [N x K]  (NT gemm), staged via LDS,
// fp32 WMMA 16x16x4. 8 waves/block; each wave owns a 32(M) x 16(N) tile
// (two accumulators sharing one B fragment -> 1.5 DS loads per WMMA,
// two independent accumulation chains). Block tile: 32 x 128.
__global__ __launch_bounds__(256)
void gemm_nt_wmma(const float* __restrict__ A1, int K1, const float* __restrict__ W1,
                  const float* __restrict__ A2, int K2, const float* __restrict__ W2,
                  const float* __restrict__ b1, const float* __restrict__ b2,
                  float* __restrict__ out, int N)
{
    __shared__ float lA[32][72];    // padded stride (288 B): float4-aligned, conflict-poor
    __shared__ float lW[128][72];

    const int tid  = threadIdx.x;
    const int lane = tid & 31;
    const int wv   = tid >> 5;      // 0..7 -> N subtile within block
    const int n0   = blockIdx.x * 128;

    v8f acc0 = {0.f, 0.f, 0.f, 0.f, 0.f, 0.f, 0.f, 0.f};
    v8f acc1 = {0.f, 0.f, 0.f, 0.f, 0.f, 0.f, 0.f, 0.f};

    for (int pass = 0; pass < 2; ++pass) {
        const float* A = pass ? A2 : A1;
        const float* W = pass ? W2 : W1;
        const int    K = pass ? K2 : K1;
        for (int kc = 0; kc < K; kc += 64) {
            // A chunk: 32 rows x 64 k -> 512 float4, 2 per thread (coalesced)
            #pragma unroll
            for (int it = 0; it < 2; ++it) {
                int idx = tid + it * 256;
                int r = idx >> 4, c4 = idx & 15;
                float4 v = *(const float4*)(A + (long)r * K + kc + c4 * 4);
                *(float4*)&lA[r][c4 * 4] = v;
            }
            // W chunk: 128 rows x 64 k -> 2048 float4, 8 per thread (coalesced)
            #pragma unroll
            for (int it = 0; it < 8; ++it) {
                int idx = tid + it * 256;
                int r = idx >> 4, c4 = idx & 15;
                float4 v = *(const float4*)(W + (long)(n0 + r) * K + kc + c4 * 4);
                *(float4*)&lW[r][c4 * 4] = v;
            }
            __syncthreads();

            // fp32 WMMA A layout: lanes 0-15 hold K+0,K+1; lanes 16-31 hold K+2,K+3
            const int hs  = (lane >= 16) ? 2 : 0;
            const int l15 = lane & 15;
            const int wr  = wv * 16 + l15;
            #pragma unroll
            for (int k4 = 0; k4 < 64; k4 += 4) {
                v2f a0 = *(const v2f*)&lA[l15][k4 + hs];
                v2f a1 = *(const v2f*)&lA[16 + l15][k4 + hs];
                v2f bf = *(const v2f*)&lW[wr][k4 + hs];
                acc0 = __builtin_amdgcn_wmma_f32_16x16x4_f32(
                         false, a0, false, bf, (short)0, acc0, false, false);
                acc1 = __builtin_amdgcn_wmma_f32_16x16x4_f32(
                         false, a1, false, bf, (short)0, acc1, false, false);
            }
            __syncthreads();
        }
    }

    // C/D layout: VGPR r, lanes 0-15 -> M=r, N=lane ; lanes 16-31 -> M=8+r, N=lane-16
    const int nn = n0 + wv * 16 + (lane & 15);
    const int mb = (lane >= 16) ? 8 : 0;
    float bias = b1[nn] + (b2 ? b2[nn] : 0.f);
    #pragma unroll
    for (int r = 0; r < 8; ++r) {
        out[(long)(mb + r) * N + nn]      = acc0[r] + bias;   // rows 0..15
        out[(long)(16 + mb + r) * N + nn] = acc1[r] + bias;   // rows 16..31
    }
}

// ---------------- LSTM pointwise: h,c <- cell(gates, h, c) ----------------
__global__ void lstm_pointwise(const float* __restrict__ gates,
                               float* __restrict__ h, float* __restrict__ c) {
    int i = blockIdx.x * blockDim.x + threadIdx.x;   // 0 .. 32*512-1
    int m = i >> 9, j = i & 511;
    const float* g = gates + (long)m * G_;
    float ig = sigmoidf_(g[j]);
    float fg = sigmoidf_(g[512 + j]);
    float gg = tanhf(g[1024 + j]);
    float og = sigmoidf_(g[1536 + j]);
    float cn = fg * c[i] + ig * gg;
    c[i] = cn;
    h[i] = og * tanhf(cn);
}

// -------- per-row log_softmax write + greedy argmax (first-index ties) --------
__global__ __launch_bounds__(256)
void logsoftmax_argmax(const float* __restrict__ logits, float* __restrict__ out,
                       int* __restrict__ tok, int t) {
    __shared__ float sval[256];
    __shared__ int   sidx[256];
    __shared__ float ssum[256];
    const int b   = blockIdx.x;
    const int tid = threadIdx.x;
    const float* row = logits + (long)b * V_;

    float m = -INFINITY; int mi = 0x7fffffff;
    for (int v = tid; v < V_; v += 256) {
        float x = row[v];
        if (x > m) { m = x; mi = v; }
    }
    sval[tid] = m; sidx[tid] = mi;
    __syncthreads();
    for (int s = 128; s > 0; s >>= 1) {
        if (tid < s) {
            float o = sval[tid + s]; int oi = sidx[tid + s];
            if (o > sval[tid] || (o == sval[tid] && oi < sidx[tid])) {
                sval[tid] = o; sidx[tid] = oi;
            }
        }
        __syncthreads();
    }
    const float rmax = sval[0];

    float acc = 0.f;
    for (int v = tid; v < V_; v += 256) acc += expf(row[v] - rmax);
    ssum[tid] = acc;
    __syncthreads();
    for (int s = 128; s > 0; s >>= 1) {
        if (tid < s) ssum[tid] += ssum[tid + s];
        __syncthreads();
    }
    const float logZ = rmax + logf(ssum[0]);

    float* orow = out + ((long)b * TMAX + t) * V_;   // out[b][t][:]
    for (int v = tid; v < V_; v += 256) orow[v] = row[v] - logZ;
    if (tid == 0) tok[b] = sidx[0];
}

extern "C" void kernel_launch(void* const* d_in, const int* in_sizes, int n_in,
                              void* d_out, int out_size, void* d_ws, size_t ws_size,
                              hipStream_t stream) {
    (void)in_sizes; (void)n_in; (void)out_size; (void)ws_size;
    const float* ctx  = (const float*)d_in[0];
    const float* emb  = (const float*)d_in[1];
    const float* Wih0 = (const float*)d_in[2];
    const float* Whh0 = (const float*)d_in[3];
    const float* bih0 = (const float*)d_in[4];
    const float* bhh0 = (const float*)d_in[5];
    const float* Wih1 = (const float*)d_in[6];
    const float* Whh1 = (const float*)d_in[7];
    const float* bih1 = (const float*)d_in[8];
    const float* bhh1 = (const float*)d_in[9];
    const float* Wout = (const float*)d_in[10];
    const float* bout = (const float*)d_in[11];
    float* out = (float*)d_out;

    float* ws     = (float*)d_ws;
    float* h0     = ws;                  // 32*512
    float* c0     = h0 + B_ * H_;
    float* h1     = c0 + B_ * H_;
    float* c1     = h1 + B_ * H_;
    float* x0     = c1 + B_ * H_;        // 32*1024
    float* gates  = x0 + B_ * 1024;      // 32*2048
    float* logits = gates + B_ * G_;     // 32*32000
    int*   tok    = (int*)(logits + (long)B_ * V_);

    init_state<<<64, 256, 0, stream>>>(h0, c0, h1, c1, tok);

    for (int t = 0; t < TMAX; ++t) {
        embed_concat<<<128, 256, 0, stream>>>(emb, ctx, tok, x0);
        // layer 0: gates = x0*Wih0^T + h0*Whh0^T + b      (K = 1024 + 512, N = 2048)
        gemm_nt_wmma<<<G_ / 128, 256, 0, stream>>>(x0, 1024, Wih0, h0, 512, Whh0,
                                                   bih0, bhh0, gates, G_);
        lstm_pointwise<<<64, 256, 0, stream>>>(gates, h0, c0);
        // layer 1: gates = h0*Wih1^T + h1*Whh1^T + b      (K = 512 + 512, N = 2048)
        gemm_nt_wmma<<<G_ / 128, 256, 0, stream>>>(h0, 512, Wih1, h1, 512, Whh1,
                                                   bih1, bhh1, gates, G_);
        lstm_pointwise<<<64, 256, 0, stream>>>(gates, h1, c1);
        // logits = h1*Wout^T + bout                       (K = 512, N = 32000)
        gemm_nt_wmma<<<V_ / 128, 256, 0, stream>>>(h1, 512, Wout, nullptr, 0, nullptr,
                                                   bout, nullptr, logits, V_);
        logsoftmax_argmax<<<B_, 256, 0, stream>>>(logits, out, tok, t);
    }
}